// GATModel_48172353192008
// MI455X (gfx1250) — compile-verified
//
#include <hip/hip_runtime.h>

#define NNODES 50000
#define NEDGES 1600000
#define ETOT   (NNODES + NEDGES)   // edges + self loops
#define NEG_SLOPE 0.2f

typedef __attribute__((ext_vector_type(2))) float v2f;
typedef __attribute__((ext_vector_type(8))) float v8f;

// ---- order-preserving float<->uint encoding for atomicMax-based segment max ----
__device__ __forceinline__ unsigned fenc(float f) {
  unsigned u = __float_as_uint(f);
  return (u & 0x80000000u) ? ~u : (u | 0x80000000u);
}
__device__ __forceinline__ float fdec(unsigned u) {
  unsigned b = (u & 0x80000000u) ? (u & 0x7fffffffu) : ~u;
  return __uint_as_float(b);
}

// =====================================================================
// Generic fp32 WMMA GEMM: C[M,N] = A[M,K] @ B[K,N] (+ bias[N] if given)
// One wave32 per 16x16 output tile; V_WMMA_F32_16X16X4_F32, K stepped by 4.
// A 16x4 layout : lane = M row (lane&15); lanes16-31 hold K=2,3 (2 VGPRs)
// B 4x16 layout : lane = N col (lane&15); lanes16-31 hold K=2,3
// C/D layout    : VGPR i -> row i (lanes0-15) / row i+8 (lanes16-31)
// =====================================================================
__global__ void k_gemm_f32_wmma(const float* __restrict__ A,
                                const float* __restrict__ B,
                                const float* __restrict__ bias,
                                float* __restrict__ C,
                                int M, int K, int N) {
  const int lane = threadIdx.x & 31;
  const int wave = blockIdx.x * (blockDim.x >> 5) + (threadIdx.x >> 5);
  const int tilesN = N >> 4;
  const int tilesM = M >> 4;
  if (wave >= tilesM * tilesN) return;       // wave-uniform: EXEC all-1s inside
  const int tm = wave / tilesN, tn = wave % tilesN;
  const int m0 = tm << 4, n0 = tn << 4;
  const int r  = lane & 15;                  // row of A / col of B
  const int kh = lane >> 4;                  // K-half select (K+0,1 vs K+2,3)

  v8f acc = {};
  const float* Arow = A + (size_t)(m0 + r) * K + 2 * kh;
  const float* Bcol = B + n0 + r;
  for (int k = 0; k < K; k += 4) {
    v2f a = *reinterpret_cast<const v2f*>(Arow + k);   // A[r][k+2kh .. +1]
    v2f b;
    b.x = Bcol[(size_t)(k + 2 * kh) * N];              // B[k+2kh][col]
    b.y = Bcol[(size_t)(k + 2 * kh + 1) * N];
    acc = __builtin_amdgcn_wmma_f32_16x16x4_f32(
        false, a, false, b, (short)0, acc, false, false);
  }
#pragma unroll
  for (int i = 0; i < 8; ++i) {
    const int row = m0 + i + 8 * kh;
    const int col = n0 + r;
    float v = acc[i];
    if (bias) v += bias[col];
    C[(size_t)row * N + col] = v;
  }
}

// ---- per-(node,head) attention coefficients: asrc = <h, a_src>, adst = <h, a_dst>
__global__ void k_attn_coeff(const float* __restrict__ h,
                             const float* __restrict__ a_src,
                             const float* __restrict__ a_dst,
                             float* __restrict__ asrc, float* __restrict__ adst,
                             int n, int H, int Cc) {
  int t = blockIdx.x * blockDim.x + threadIdx.x;
  if (t >= n * H) return;
  int node = t / H, hd = t % H;
  const float* hp = h + (size_t)node * H * Cc + (size_t)hd * Cc;
  const float* as = a_src + hd * Cc;
  const float* ad = a_dst + hd * Cc;
  float s0 = 0.f, s1 = 0.f;
  for (int c = 0; c < Cc; ++c) { float v = hp[c]; s0 += v * as[c]; s1 += v * ad[c]; }
  asrc[t] = s0;
  adst[t] = s1;
}

__global__ void k_fill_f32(float* __restrict__ p, float v, long long n) {
  long long i = (long long)blockIdx.x * blockDim.x + threadIdx.x;
  if (i < n) p[i] = v;
}
__global__ void k_fill_u32(unsigned* __restrict__ p, unsigned v, long long n) {
  long long i = (long long)blockIdx.x * blockDim.x + threadIdx.x;
  if (i < n) p[i] = v;
}

// ---- pass 1: edge logits + segment max over dst (atomicMax on encoded floats)
__global__ void k_edge_logits(const int* __restrict__ ei,
                              const float* __restrict__ asrc,
                              const float* __restrict__ adst,
                              float* __restrict__ ebuf,
                              unsigned* __restrict__ emax, int H) {
  int t = blockIdx.x * blockDim.x + threadIdx.x;
  if (t >= ETOT) return;
  int s = (t < NEDGES) ? ei[t]          : (t - NEDGES);
  int d = (t < NEDGES) ? ei[NEDGES + t] : (t - NEDGES);
  for (int h = 0; h < H; ++h) {
    float e = asrc[s * H + h] + adst[d * H + h];
    e = (e > 0.f) ? e : NEG_SLOPE * e;            // LeakyReLU
    ebuf[(size_t)t * H + h] = e;
    atomicMax(&emax[d * H + h], fenc(e));
  }
}

// ---- pass 2: p = exp(e - max), segment-sum denominators
__global__ void k_edge_expsum(const int* __restrict__ ei,
                              float* __restrict__ ebuf,
                              const unsigned* __restrict__ emax,
                              float* __restrict__ denom, int H) {
  int t = blockIdx.x * blockDim.x + threadIdx.x;
  if (t >= ETOT) return;
  int d = (t < NEDGES) ? ei[NEDGES + t] : (t - NEDGES);
  for (int h = 0; h < H; ++h) {
    float p = expf(ebuf[(size_t)t * H + h] - fdec(emax[d * H + h]));
    ebuf[(size_t)t * H + h] = p;
    atomicAdd(&denom[d * H + h], p);
  }
}

// ---- pass 3: agg[dst] += alpha * h[src]  (one thread per edge x 4 channels;
//      gather/scatter working set fits in 192MB L2 -> atomics resolve in L2)
__global__ void k_edge_aggregate(const int* __restrict__ ei,
                                 const float* __restrict__ hfeat,
                                 const float* __restrict__ ebuf,
                                 const float* __restrict__ denom,
                                 float* __restrict__ agg, int H, int Cc) {
  const int HC4 = (H * Cc) >> 2;
  long long tid = (long long)blockIdx.x * blockDim.x + threadIdx.x;
  if (tid >= (long long)ETOT * HC4) return;
  int t = (int)(tid / HC4);
  int j = (int)(tid % HC4);
  int s = (t < NEDGES) ? ei[t]          : (t - NEDGES);
  int d = (t < NEDGES) ? ei[NEDGES + t] : (t - NEDGES);
  int c4 = j << 2;
  int h  = c4 / Cc;
  float alpha = ebuf[(size_t)t * H + h] / (denom[d * H + h] + 1e-16f);
  const float4 v = *reinterpret_cast<const float4*>(hfeat + (size_t)s * H * Cc + c4);
  float* o = agg + (size_t)d * H * Cc + c4;
  atomicAdd(o + 0, alpha * v.x);
  atomicAdd(o + 1, alpha * v.y);
  atomicAdd(o + 2, alpha * v.z);
  atomicAdd(o + 3, alpha * v.w);
}

// ---- bias + ReLU (in place)
__global__ void k_bias_relu(float* __restrict__ p, const float* __restrict__ bias,
                            long long n, int F) {
  long long i = (long long)blockIdx.x * blockDim.x + threadIdx.x;
  if (i >= n) return;
  float v = p[i] + bias[(int)(i % F)];
  p[i] = v > 0.f ? v : 0.f;
}

static inline int nb(long long n, int bs) { return (int)((n + bs - 1) / bs); }

extern "C" void kernel_launch(void* const* d_in, const int* in_sizes, int n_in,
                              void* d_out, int out_size, void* d_ws, size_t ws_size,
                              hipStream_t stream) {
  (void)in_sizes; (void)n_in; (void)out_size; (void)ws_size;
  const float* x      = (const float*)d_in[0];
  const int*   ei     = (const int*)  d_in[1];
  const float* W1     = (const float*)d_in[2];
  const float* a_src1 = (const float*)d_in[3];
  const float* a_dst1 = (const float*)d_in[4];
  const float* b1     = (const float*)d_in[5];
  const float* W2     = (const float*)d_in[6];
  const float* a_src2 = (const float*)d_in[7];
  const float* a_dst2 = (const float*)d_in[8];
  const float* b2     = (const float*)d_in[9];
  const float* Wl     = (const float*)d_in[10];
  const float* bl     = (const float*)d_in[11];
  float* out = (float*)d_out;

  // workspace layout (floats): h1 | agg1 | ebuf | asrc | adst | denom | emax
  float* ws   = (float*)d_ws;
  float* h1   = ws;                                   // N*128
  float* agg1 = h1   + (size_t)NNODES * 128;          // N*128 (-> h1_act)
  float* ebuf = agg1 + (size_t)NNODES * 128;          // ETOT*2
  float* asrc = ebuf + (size_t)ETOT * 2;              // N*2
  float* adst = asrc + (size_t)NNODES * 2;            // N*2
  float* denom= adst + (size_t)NNODES * 2;            // N*2
  unsigned* emax = (unsigned*)(denom + (size_t)NNODES * 2); // N*2
  float* h2   = h1;                                   // reuse: N*64
  float* agg2 = h1 + (size_t)NNODES * 64;             // reuse: N*64

  const int BS = 256;

  // ================= layer 1 (heads=2, ch=64) =================
  {
    int tiles = (NNODES / 16) * (128 / 16);          // 25000 waves
    k_gemm_f32_wmma<<<nb(tiles, 8), BS, 0, stream>>>(x, W1, nullptr, h1,
                                                     NNODES, 128, 128);
    k_attn_coeff<<<nb((long long)NNODES * 2, BS), BS, 0, stream>>>(
        h1, a_src1, a_dst1, asrc, adst, NNODES, 2, 64);
    k_fill_f32<<<nb((long long)NNODES * 128, BS), BS, 0, stream>>>(agg1, 0.f, (long long)NNODES * 128);
    k_fill_f32<<<nb((long long)NNODES * 2, BS), BS, 0, stream>>>(denom, 0.f, (long long)NNODES * 2);
    k_fill_u32<<<nb((long long)NNODES * 2, BS), BS, 0, stream>>>(emax, 0u, (long long)NNODES * 2);
    k_edge_logits<<<nb(ETOT, BS), BS, 0, stream>>>(ei, asrc, adst, ebuf, emax, 2);
    k_edge_expsum<<<nb(ETOT, BS), BS, 0, stream>>>(ei, ebuf, emax, denom, 2);
    k_edge_aggregate<<<nb((long long)ETOT * 32, BS), BS, 0, stream>>>(
        ei, h1, ebuf, denom, agg1, 2, 64);
    k_bias_relu<<<nb((long long)NNODES * 128, BS), BS, 0, stream>>>(
        agg1, b1, (long long)NNODES * 128, 128);
  }

  // ================= layer 2 (heads=1, ch=64) =================
  {
    int tiles = (NNODES / 16) * (64 / 16);           // 12500 waves
    k_gemm_f32_wmma<<<nb(tiles, 8), BS, 0, stream>>>(agg1, W2, nullptr, h2,
                                                     NNODES, 128, 64);
    k_attn_coeff<<<nb((long long)NNODES, BS), BS, 0, stream>>>(
        h2, a_src2, a_dst2, asrc, adst, NNODES, 1, 64);
    k_fill_f32<<<nb((long long)NNODES * 64, BS), BS, 0, stream>>>(agg2, 0.f, (long long)NNODES * 64);
    k_fill_f32<<<nb((long long)NNODES, BS), BS, 0, stream>>>(denom, 0.f, (long long)NNODES);
    k_fill_u32<<<nb((long long)NNODES, BS), BS, 0, stream>>>(emax, 0u, (long long)NNODES);
    k_edge_logits<<<nb(ETOT, BS), BS, 0, stream>>>(ei, asrc, adst, ebuf, emax, 1);
    k_edge_expsum<<<nb(ETOT, BS), BS, 0, stream>>>(ei, ebuf, emax, denom, 1);
    k_edge_aggregate<<<nb((long long)ETOT * 16, BS), BS, 0, stream>>>(
        ei, h2, ebuf, denom, agg2, 1, 64);
    k_bias_relu<<<nb((long long)NNODES * 64, BS), BS, 0, stream>>>(
        agg2, b2, (long long)NNODES * 64, 64);
  }

  // ================= final linear: out = h2_act @ Wl + bl =================
  {
    int tiles = (NNODES / 16) * (16 / 16);           // 3125 waves
    k_gemm_f32_wmma<<<nb(tiles, 8), BS, 0, stream>>>(agg2, Wl, bl, out,
                                                     NNODES, 64, 16);
  }
}